// GCN_1649267442136
// MI455X (gfx1250) — compile-verified
//
#include <hip/hip_runtime.h>

// ---------------------------------------------------------------------------
// GCN layer for MI455X (gfx1250, wave32).
//   h1  = relu( Agg( x @ W1 ) + b1 )
//   out =       Agg( h1 @ W2 ) + b2
// GEMMs: V_WMMA_F32_16X16X4_F32 fed from LDS tiles staged with
//        GLOBAL_LOAD_ASYNC_TO_LDS_B128 (ASYNCcnt) when available.
// Aggregation: norm precomputed per edge; one edge per wave32;
//        global_atomic_add_f32 scatter (51MB accumulator lives in 192MB L2).
// ---------------------------------------------------------------------------

typedef float v2f __attribute__((ext_vector_type(2)));
typedef float v8f __attribute__((ext_vector_type(8)));
typedef int   v4i __attribute__((vector_size(16)));   // matches builtin param

#if defined(__has_builtin)
#if __has_builtin(__builtin_amdgcn_global_load_async_to_lds_b128) && \
    __has_builtin(__builtin_amdgcn_s_wait_asynccnt)
#define HAVE_ASYNC_LDS 1
#endif
#endif
#ifndef HAVE_ASYNC_LDS
#define HAVE_ASYNC_LDS 0
#endif

#if HAVE_ASYNC_LDS
typedef __attribute__((address_space(1))) v4i* g1_v4i_p;   // global (64-bit)
typedef __attribute__((address_space(3))) v4i* l3_v4i_p;   // LDS (32-bit)
#endif

// 16-byte global->LDS copy. Async DMA path on gfx1250, vector copy fallback.
__device__ __forceinline__ void stage16(const float* g, float* s) {
#if HAVE_ASYNC_LDS
    __builtin_amdgcn_global_load_async_to_lds_b128(
        (g1_v4i_p)(uintptr_t)g,
        (l3_v4i_p)(unsigned)(uintptr_t)s,
        /*offset=*/0, /*cpol=*/0);
#else
    *(float4*)s = *(const float4*)g;
#endif
}

__device__ __forceinline__ void wait_stage() {
#if HAVE_ASYNC_LDS
    __builtin_amdgcn_s_wait_asynccnt(0);   // async LDS writes not covered by barrier waits
#endif
}

// ---------------- degree / normalization ----------------

__global__ void __launch_bounds__(256)
init_deg(float* __restrict__ deg, int n) {
    int i = blockIdx.x * 256 + threadIdx.x;
    if (i < n) deg[i] = 1.0f;              // self-loop weight
}

__global__ void __launch_bounds__(256)
accum_deg(const int* __restrict__ dst, const float* __restrict__ w,
          float* __restrict__ deg, int E) {
    int e = blockIdx.x * 256 + threadIdx.x;
    if (e < E) atomicAdd(&deg[dst[e]], w[e]);
}

__global__ void __launch_bounds__(256)
make_dinv(float* __restrict__ deg, int n) {
    int i = blockIdx.x * 256 + threadIdx.x;
    if (i < n) {
        float d = deg[i];
        deg[i] = (d > 0.0f) ? rsqrtf(d) : 0.0f;   // deg -> D^{-1/2} in place
    }
}

__global__ void __launch_bounds__(256)
compute_norm(const int* __restrict__ src, const int* __restrict__ dst,
             const float* __restrict__ w, const float* __restrict__ dinv,
             float* __restrict__ nrm, int E) {
    int e = blockIdx.x * 256 + threadIdx.x;
    if (e < E) nrm[e] = dinv[src[e]] * w[e] * dinv[dst[e]];
}

// ---------------- fp32 WMMA GEMM with LDS staging ----------------
// C[M x Nc] = A[M x K] @ B[K x Nc].  K, Nc multiples of 64/16.
// Block: 256 thr = 8 waves -> 128 rows x 16 cols. K chunked by 64.

#define TM   128            // rows per block
#define KC   64             // K chunk
#define AP   (KC + 4)       // padded LDS row stride (68): 16B-aligned rows,
                            // conflict-free ds_load_b64 (stride 68 % 64 == 4)

__global__ void __launch_bounds__(256)
gemm16_wmma_f32(const float* __restrict__ A, const float* __restrict__ B,
                float* __restrict__ C, int M, int K, int Nc) {
    __shared__ float lds_a[TM * AP];    // ~34.0 KB
    __shared__ float lds_b[KC * 16];    //   4.0 KB

    const int tid  = threadIdx.x;
    const int lane = tid & 31;
    const int wave = tid >> 5;
    const int half = lane >> 4;
    const int l    = lane & 15;

    const int n0     = blockIdx.x << 4;
    const int m_base = blockIdx.y * TM;
    const int m0     = m_base + (wave << 4);
    const bool wvalid = (m0 + 16 <= M);

    v8f acc = {0.f, 0.f, 0.f, 0.f, 0.f, 0.f, 0.f, 0.f};

    for (int kc = 0; kc < K; kc += KC) {
        // stage A chunk: TM x KC, 16B ops, 8 per thread, coalesced along K
#pragma unroll
        for (int c = tid; c < TM * (KC / 4); c += 256) {
            const int r = c >> 4;                 // KC/4 == 16 chunks per row
            const int p = c & 15;
            if (m_base + r < M)
                stage16(A + (size_t)(m_base + r) * K + kc + p * 4,
                        &lds_a[r * AP + p * 4]);
        }
        // stage B chunk: KC x 16, one 16B op per thread
        {
            const int r = tid >> 2;
            const int p = tid & 3;
            stage16(B + (size_t)(kc + r) * Nc + n0 + p * 4,
                    &lds_b[r * 16 + p * 4]);
        }
        wait_stage();
        __syncthreads();

        if (wvalid) {
            const float* ar = &lds_a[(wave * 16 + l) * AP];
#pragma unroll
            for (int k0 = 0; k0 < KC; k0 += 4) {
                const int ka = k0 + 2 * half;
                // A 16x4: lanes 0-15 K={k0,k0+1}, lanes 16-31 K={k0+2,k0+3}
                v2f a = *(const v2f*)(ar + ka);
                // B 4x16: V0 = rows {k0, k0+2}, V1 = rows {k0+1, k0+3}
                v2f b;
                b.x = lds_b[ka * 16 + l];
                b.y = lds_b[(ka + 1) * 16 + l];
                acc = __builtin_amdgcn_wmma_f32_16x16x4_f32(
                        false, a, false, b, (short)0, acc, false, false);
            }
        }
        __syncthreads();                          // protect LDS reuse
    }

    if (wvalid) {
        // C 16x16: VGPR r -> row (r + 8*half), col n0+l
        float* crow = C + (size_t)(m0 + 8 * half) * Nc + n0 + l;
#pragma unroll
        for (int r = 0; r < 8; ++r) crow[(size_t)r * Nc] = acc[r];
    }
}

// ---------------- scatter aggregation ----------------

__global__ void __launch_bounds__(256)
zero_f32(float* __restrict__ p, int n) {
    int i = blockIdx.x * 256 + threadIdx.x;
    if (i < n) p[i] = 0.0f;
}

// One edge per wave32: per-edge scalars fetched once per wave (HW broadcast),
// each lane handles float4 chunks of the feature row.
__global__ void __launch_bounds__(256)
edge_aggregate(const int* __restrict__ src, const int* __restrict__ dst,
               const float* __restrict__ nrm, const float* __restrict__ h,
               float* __restrict__ agg, int E, int C) {
    const int e    = (blockIdx.x * 256 + threadIdx.x) >> 5;
    const int lane = threadIdx.x & 31;
    if (e >= E) return;

    const int s = src[e];
    const int d = dst[e];
    const float nv = nrm[e];

    const float* hp = h + (size_t)s * C;
    float* ap = agg + (size_t)d * C;
    for (int c = lane * 4; c < C; c += 128) {
        const float4 hv = *(const float4*)(hp + c);
        atomicAdd(ap + c + 0, hv.x * nv);
        atomicAdd(ap + c + 1, hv.y * nv);
        atomicAdd(ap + c + 2, hv.z * nv);
        atomicAdd(ap + c + 3, hv.w * nv);
    }
}

// agg[i] += h[i] * dinv[row]^2 (self-loop, weight 1) + bias[col]; optional relu.
__global__ void __launch_bounds__(256)
finish_node(const float* __restrict__ h, const float* __restrict__ dinv,
            const float* __restrict__ bias, float* __restrict__ agg,
            int total, int lgC, int do_relu) {
    int i = blockIdx.x * 256 + threadIdx.x;
    if (i >= total) return;
    const int row = i >> lgC;
    const int col = i & ((1 << lgC) - 1);
    const float di = dinv[row];
    float v = agg[i] + h[i] * di * di + bias[col];
    agg[i] = do_relu ? fmaxf(v, 0.0f) : v;
}

// ---------------------------------------------------------------------------

extern "C" void kernel_launch(void* const* d_in, const int* in_sizes, int n_in,
                              void* d_out, int out_size, void* d_ws, size_t ws_size,
                              hipStream_t stream) {
    const float* x  = (const float*)d_in[0];
    const int*   ei = (const int*)  d_in[1];
    const float* ew = (const float*)d_in[2];
    const float* W1 = (const float*)d_in[3];
    const float* b1 = (const float*)d_in[4];
    const float* W2 = (const float*)d_in[5];
    const float* b2 = (const float*)d_in[6];

    const int E  = in_sizes[2];
    const int H  = in_sizes[4];            // 256
    const int F  = in_sizes[3] / H;        // 512
    const int Nn = in_sizes[0] / F;        // 50000
    const int O  = in_sizes[6];            // 128

    const int* srcIdx = ei;
    const int* dstIdx = ei + E;

    // workspace layout (256B-aligned)
    char* wsb = (char*)d_ws;
    size_t off = 0;
    float* deg  = (float*)(wsb + off);
    off += ((size_t)Nn * sizeof(float) + 255) & ~(size_t)255;
    float* nrm  = (float*)(wsb + off);
    off += ((size_t)E * sizeof(float) + 255) & ~(size_t)255;
    float* h1   = (float*)(wsb + off);  off += (size_t)Nn * H * sizeof(float);
    float* agg1 = (float*)(wsb + off);  off += (size_t)Nn * H * sizeof(float);
    float* h2   = (float*)(wsb + off);
    float* agg2 = (float*)d_out;

    const int TB = 256;
    const int lgH = 31 - __builtin_clz((unsigned)H);
    const int lgO = 31 - __builtin_clz((unsigned)O);

    // --- symmetric normalization D^{-1/2}, per-edge norm ---
    init_deg    <<<(Nn + TB - 1) / TB, TB, 0, stream>>>(deg, Nn);
    accum_deg   <<<(E  + TB - 1) / TB, TB, 0, stream>>>(dstIdx, ew, deg, E);
    make_dinv   <<<(Nn + TB - 1) / TB, TB, 0, stream>>>(deg, Nn);
    compute_norm<<<(E  + TB - 1) / TB, TB, 0, stream>>>(srcIdx, dstIdx, ew, deg, nrm, E);

    // --- conv1: h1 = x @ W1 ---
    gemm16_wmma_f32<<<dim3(H / 16, (Nn + TM - 1) / TM), TB, 0, stream>>>(
        x, W1, h1, Nn, F, H);

    const int n1 = Nn * H;
    zero_f32<<<(n1 + TB - 1) / TB, TB, 0, stream>>>(agg1, n1);
    edge_aggregate<<<(E + 7) / 8, TB, 0, stream>>>(srcIdx, dstIdx, nrm, h1, agg1, E, H);
    finish_node<<<(n1 + TB - 1) / TB, TB, 0, stream>>>(h1, deg, b1, agg1, n1, lgH, 1);

    // --- conv2: h2 = relu_agg1 @ W2 ---
    gemm16_wmma_f32<<<dim3(O / 16, (Nn + TM - 1) / TM), TB, 0, stream>>>(
        agg1, W2, h2, Nn, H, O);

    const int n2 = Nn * O;
    zero_f32<<<(n2 + TB - 1) / TB, TB, 0, stream>>>(agg2, n2);
    edge_aggregate<<<(E + 7) / 8, TB, 0, stream>>>(srcIdx, dstIdx, nrm, h2, agg2, E, O);
    finish_node<<<(n2 + TB - 1) / TB, TB, 0, stream>>>(h2, deg, b2, agg2, n2, lgO, 0);
}